// AdvancedWaveletDecomp_80985903333574
// MI455X (gfx1250) — compile-verified
//
#include <hip/hip_runtime.h>
#include <math.h>

// ---------------------------------------------------------------------------
// AdvancedWaveletDecomp for MI455X (gfx1250, wave32).
// Core compute: C=64->64 K=3 convs mapped onto V_WMMA_F32_16X16X4_F32
// (fp32 WMMA keeps full reference precision through 16 recursive RK4 evals).
// B operand is linearized into LDS ([row-pair][col][2] layout, 96-dword
// q-stride) so the fully-unrolled 48-chunk GEMM loop runs with pure
// immediate-offset ds_load_b64 / global_load_b64 feeding each WMMA.
// ---------------------------------------------------------------------------

#define CB   8
#define CC   64
#define CL   2048
#define CHID 64
#define CFL  7
#define NLEV 3
#define NSTEP 4          // T-1 RK4 steps
#define NTD  16
#define NPER (CC*CL)     // 131072 = 2^17
#define NTOT (CB*CC*CL)  // 1048576
#define KR   192         // GEMM K = 64 chans * 3 taps
#define XQ   96          // dword stride per row-pair in Xr2 (bank-conflict free)

typedef float v2f __attribute__((ext_vector_type(2)));
typedef float v8f __attribute__((ext_vector_type(8)));

#if defined(__gfx1250__) && __has_builtin(__builtin_amdgcn_wmma_f32_16x16x4_f32)
#define USE_WMMA_F32 1
#endif

__device__ __forceinline__ float gelu_f(float x){ return 0.5f*x*(1.0f+erff(x*0.70710678118654752f)); }
__device__ __forceinline__ float sigmoid_f(float x){ return 1.0f/(1.0f+expf(-x)); }
__device__ __forceinline__ float silu_f(float x){ return x/(1.0f+expf(-x)); }

// ---------------------------------------------------------------- row mean
__global__ void row_mean_kernel(const float* __restrict__ in, float* __restrict__ stat){
  int row = blockIdx.x;
  const float* p = in + (size_t)row*CL;
  float s = 0.f;
  for(int i=threadIdx.x;i<CL;i+=256) s += p[i];
  __shared__ float red[256];
  red[threadIdx.x]=s; __syncthreads();
  for(int off=128;off>0;off>>=1){ if(threadIdx.x<off) red[threadIdx.x]+=red[threadIdx.x+off]; __syncthreads(); }
  if(threadIdx.x==0) stat[row]=red[0]*(1.0f/CL);
}

// -------------------------------------------------- dywan MLP (+ ortho loss)
__global__ void dywan_kernel(const float* __restrict__ stat,
    const float* __restrict__ w1,const float* __restrict__ b1,
    const float* __restrict__ wg1,const float* __restrict__ bg1,
    const float* __restrict__ wg2,const float* __restrict__ bg2,
    float* __restrict__ lo,float* __restrict__ hi,float* __restrict__ ortho){
  __shared__ float sh1[CB][CHID];
  __shared__ float sh2[CB][2*CHID];
  __shared__ float slo[CB][CFL];
  int tid=threadIdx.x; int b=tid>>5; int lane=tid&31;
  for(int o=lane;o<CHID;o+=32){
    float a=b1[o];
    for(int c=0;c<CC;c++) a += stat[b*CC+c]*w1[o*CC+c];
    sh1[b][o]=gelu_f(a);
  }
  __syncthreads();
  for(int o=lane;o<2*CHID;o+=32){
    float a=bg1[o];
    for(int c=0;c<CHID;c++) a += sh1[b][c]*wg1[o*CHID+c];
    sh2[b][o]=gelu_f(a);
  }
  __syncthreads();
  if(lane<2*CFL){
    float a=bg2[lane];
    for(int c=0;c<2*CHID;c++) a += sh2[b][c]*wg2[lane*2*CHID+c];
    if(lane<CFL){ slo[b][lane]=a; lo[b*CFL+lane]=a; }
    else hi[b*CFL+(lane-CFL)]=a;
  }
  __syncthreads();
  if(tid==0){
    float smooth=0.f, shift=0.f, amp=0.f;
    for(int bb=0;bb<CB;bb++){
      float prev=0.f, n2=0.f;
      for(int j=0;j<CFL;j++){
        float v=slo[bb][j];
        smooth += fabsf(v-prev); prev=v;
        n2 += v*v;
      }
      smooth += fabsf(prev);              // trailing pad diff
      float nrm=sqrtf(n2), den=nrm+1e-8f;
      float asum=0.f;
      for(int j=0;j<CFL;j++) asum += fabsf(slo[bb][j])/den;
      shift += asum*asum;                 // sum_ij |lo_n_i*sh_j| == A_b^2 for every roll
      amp += fabsf(n2/(den*den)-1.0f);
    }
    float lo_smooth = smooth/(float)(CB*(CFL+1));
    float shift_loss = 3.0f*(shift/(float)CB)/(float)(CFL*CFL);
    float amp_loss = amp/(float)CB;
    ortho[0] = 0.01f*(shift_loss+amp_loss) + lo_smooth*0.1f;
  }
}

// -------------------------------- per-(b,c) 7-tap depthwise conv, edge pad
__global__ void filter_kernel(const float* __restrict__ ap,
    const float* __restrict__ lo,const float* __restrict__ hi,
    float* __restrict__ outA,float* __restrict__ outD){
  __shared__ float s[CL+6];
  int row=blockIdx.x; int b=row>>6;
  const float* p=ap+(size_t)row*CL;
  for(int i=threadIdx.x;i<CL+6;i+=256){
    int q=i-3; q = q<0?0:(q>CL-1?CL-1:q);   // edge replicate
    s[i]=p[q];
  }
  __syncthreads();
  float fl[CFL],fh[CFL];
  for(int t=0;t<CFL;t++){ fl[t]=lo[b*CFL+t]; fh[t]=hi[b*CFL+t]; }
  for(int x=threadIdx.x;x<CL;x+=256){
    float aA=0.f,aD=0.f;
    #pragma unroll
    for(int t=0;t<CFL;t++){ float v=s[x+t]; aA+=v*fl[t]; aD+=v*fh[t]; }
    outA[(size_t)row*CL+x]=aA; outD[(size_t)row*CL+x]=aD;
  }
}

// ------------------------------------------------------ adaptive time grid
__device__ __forceinline__ float inten_at(const float* s,int i){
  float acc=0.f;
  #pragma unroll
  for(int d=-2;d<=2;d++){
    int k=i+d;
    if(k>=0 && k<CL-1) acc += fabsf(s[k+1]-s[k]);
  }
  return acc*0.2f;
}

__global__ void tg_minmax_kernel(const float* __restrict__ sig,
    float* __restrict__ rowmin,float* __restrict__ rowmax){
  __shared__ float s[CL];
  int row=blockIdx.x;
  const float* p=sig+(size_t)row*CL;
  for(int i=threadIdx.x;i<CL;i+=256) s[i]=p[i];
  __syncthreads();
  float mn=3.4e38f, mx=-3.4e38f;
  for(int i=threadIdx.x;i<CL-1;i+=256){
    float v=inten_at(s,i);
    mn=fminf(mn,v); mx=fmaxf(mx,v);
  }
  __shared__ float rmn[256],rmx[256];
  rmn[threadIdx.x]=mn; rmx[threadIdx.x]=mx; __syncthreads();
  for(int off=128;off>0;off>>=1){
    if(threadIdx.x<off){
      rmn[threadIdx.x]=fminf(rmn[threadIdx.x],rmn[threadIdx.x+off]);
      rmx[threadIdx.x]=fmaxf(rmx[threadIdx.x],rmx[threadIdx.x+off]);
    }
    __syncthreads();
  }
  if(threadIdx.x==0){ rowmin[row]=rmn[0]; rowmax[row]=rmx[0]; }
}

__global__ void tg_global_kernel(const float* __restrict__ rowmin,
    const float* __restrict__ rowmax,float* __restrict__ gstat){
  __shared__ float rmn[256],rmx[256];
  float mn=3.4e38f,mx=-3.4e38f;
  for(int r=threadIdx.x;r<CB*CC;r+=256){ mn=fminf(mn,rowmin[r]); mx=fmaxf(mx,rowmax[r]); }
  rmn[threadIdx.x]=mn; rmx[threadIdx.x]=mx; __syncthreads();
  for(int off=128;off>0;off>>=1){
    if(threadIdx.x<off){
      rmn[threadIdx.x]=fminf(rmn[threadIdx.x],rmn[threadIdx.x+off]);
      rmx[threadIdx.x]=fmaxf(rmx[threadIdx.x],rmx[threadIdx.x+off]);
    }
    __syncthreads();
  }
  if(threadIdx.x==0){ gstat[0]=rmn[0]; gstat[1]=rmx[0]; }
}

// partial sums of td over [0,511),[0,1023),[0,1534),[0,2046) per row
__global__ void tg_sums_kernel(const float* __restrict__ sig,
    const float* __restrict__ gstat,float* __restrict__ rowsum){
  __shared__ float s[CL];
  int row=blockIdx.x;
  const float* p=sig+(size_t)row*CL;
  for(int i=threadIdx.x;i<CL;i+=256) s[i]=p[i];
  __syncthreads();
  float mn=gstat[0], mx=gstat[1];
  bool uni = (mx-mn)<1e-8f;
  float sc = 0.9f/(mx-mn+1e-30f);
  float a0=0,a1=0,a2=0,a3=0;
  for(int i=threadIdx.x;i<2046;i+=256){
    float v=inten_at(s,i);
    float td = uni ? 0.5f : ((v-mn)*sc+0.1f);
    a3+=td;
    if(i<1534) a2+=td;
    if(i<1023) a1+=td;
    if(i<511)  a0+=td;
  }
  __shared__ float red[4][256];
  red[0][threadIdx.x]=a0; red[1][threadIdx.x]=a1;
  red[2][threadIdx.x]=a2; red[3][threadIdx.x]=a3;
  __syncthreads();
  for(int off=128;off>0;off>>=1){
    if(threadIdx.x<off)
      for(int q=0;q<4;q++) red[q][threadIdx.x]+=red[q][threadIdx.x+off];
    __syncthreads();
  }
  if(threadIdx.x==0) for(int q=0;q<4;q++) rowsum[row*4+q]=red[q][0];
}

__global__ void tg_final_kernel(const float* __restrict__ rowsum,float* __restrict__ ts){
  __shared__ float red[5][256];
  float s0=0,s1=0,s2=0,s3=0,m=-3.4e38f;
  for(int r=threadIdx.x;r<CB*CC;r+=256){
    s0+=rowsum[r*4+0]; s1+=rowsum[r*4+1]; s2+=rowsum[r*4+2]; s3+=rowsum[r*4+3];
    m=fmaxf(m,rowsum[r*4+3]);
  }
  red[0][threadIdx.x]=s0; red[1][threadIdx.x]=s1; red[2][threadIdx.x]=s2;
  red[3][threadIdx.x]=s3; red[4][threadIdx.x]=m;
  __syncthreads();
  for(int off=128;off>0;off>>=1){
    if(threadIdx.x<off){
      for(int q=0;q<4;q++) red[q][threadIdx.x]+=red[q][threadIdx.x+off];
      red[4][threadIdx.x]=fmaxf(red[4][threadIdx.x],red[4][threadIdx.x+off]);
    }
    __syncthreads();
  }
  if(threadIdx.x==0){
    float M=fmaxf(red[4][0],1e-12f);
    float inv=1.0f/((float)(CB*CC)*M);
    ts[0]=0.f;
    ts[1]=red[0][0]*inv; ts[2]=red[1][0]*inv;
    ts[3]=red[2][0]*inv; ts[4]=red[3][0]*inv;
  }
}

// -------------------------------- FiLM vector: g1 = 1+gamma+se, beta
__global__ void modvec_kernel(const float* __restrict__ tsb,int j,float tfrac,
    const float* __restrict__ wt,const float* __restrict__ bt,
    const float* __restrict__ wm,const float* __restrict__ bm,
    const float* __restrict__ emb,int lvl,float* __restrict__ g1beta){
  __shared__ float temb[NTD];
  float t = tsb[j] + tfrac*(tsb[j+1]-tsb[j]);
  int tid=threadIdx.x;
  if(tid<NTD) temb[tid]=silu_f(wt[lvl*NTD+tid]*t + bt[lvl*NTD+tid]);
  __syncthreads();
  if(tid<2*CHID){
    float g=bm[lvl*2*CHID+tid];
    for(int q=0;q<NTD;q++) g += wm[(lvl*2*CHID+tid)*NTD+q]*temb[q];
    if(tid<CHID) g1beta[tid]=1.0f+g+emb[(lvl*8+lvl)*CHID+tid];
    else         g1beta[tid]=g;
  }
}

// ---------------------------------------------------------------------------
// 64->64 K=3 conv as GEMM via V_WMMA_F32_16X16X4_F32.
// Block: 256 thr (8 waves) covers b, 32 positions, all 64 out channels.
// Wave w: mt = w&3 (16 out chans), nt = w>>2 (16 positions).
// B operand pre-linearized: Xr2[q=r>>1][col][r&1], q-stride XQ=96 dwords
// -> each WMMA is fed by one immediate-offset ds_load_b64 + global_load_b64.
// Fused epilogues: GELU (ode conv1) / SiLU-FiLM-k+yacc (ode conv2) / gate.
// ---------------------------------------------------------------------------
#define MODE_GELU 0
#define MODE_ODE2 1
#define MODE_GATE 2

__global__ __launch_bounds__(256) void conv3_kernel(
    const float* __restrict__ in, const float* __restrict__ kbuf,
    const float* __restrict__ tsb, int jstep, float afrac,
    const float* __restrict__ W, const float* __restrict__ bias,
    int mode, const float* __restrict__ g1beta,
    const float* __restrict__ yin, float* __restrict__ yacc, float wfrac,
    float* __restrict__ kout, const float* __restrict__ detail,
    float* __restrict__ out)
{
  __shared__ float Xs[CC*34];
  __shared__ float Xr2[(KR/2)*XQ];       // 96 row-pairs x 96-dword stride
  int bid=blockIdx.x; int b=bid>>6; int l0=(bid&63)<<5;
  float dt=0.f, alpha=0.f;
  if(tsb){ dt = tsb[jstep+1]-tsb[jstep]; alpha = afrac*dt; }
  size_t base=(size_t)b*NPER;
  bool stageK = (mode==MODE_GELU) && (kbuf!=nullptr);
  for(int i=threadIdx.x;i<CC*34;i+=256){
    int c=i/34, p=i-34*c; int gp=l0+p-1;
    float v=0.f;
    if(gp>=0 && gp<CL){
      size_t idx=base+(size_t)c*CL+gp;
      v=in[idx];
      if(stageK) v += alpha*kbuf[idx];
    }
    Xs[i]=v;
  }
  __syncthreads();
  // linearize: Xr2[(r>>1)*XQ + 2*col + (r&1)] = X[r=(i,k)][col] = Xs[i*34+col+k]
  for(int idx=threadIdx.x; idx<KR*32; idx+=256){
    int r=idx>>5, jc=idx&31;
    int c=r/3, k=r-3*c;
    Xr2[(r>>1)*XQ + 2*jc + (r&1)] = Xs[c*34 + jc + k];
  }
  __syncthreads();

  int tid=threadIdx.x, w=tid>>5, lane=tid&31;
  int mt=w&3, nt=w>>2;
  int n=lane&15, half=lane>>4;
  int nbase=nt*16;
  float vals[8];
#if defined(USE_WMMA_F32)
  v8f acc={0.f,0.f,0.f,0.f,0.f,0.f,0.f,0.f};
  // A row m = lane&15; halves split K as {K0,K1 | K2,K3}
  const float* Wr2 = W + (size_t)(mt*16 + n)*KR + 2*half;
  const float* Bp  = &Xr2[half*XQ + 2*(nbase+n)];
  #pragma unroll
  for(int cc=0;cc<48;cc++){
    v2f a  = *(const v2f*)(Wr2 + 4*cc);       // rows K=4cc+2h, +1 (8B aligned)
    v2f bf = *(const v2f*)(Bp + 2*XQ*cc);     // same K rows, this lane's col
    acc = __builtin_amdgcn_wmma_f32_16x16x4_f32(false,a,false,bf,(short)0,acc,false,false);
  }
  #pragma unroll
  for(int r=0;r<8;r++) vals[r]=acc[r];
#else
  for(int r=0;r<8;r++){
    int mloc=r+8*half;
    const float* Wr=W+(size_t)(mt*16+mloc)*KR;
    float a=0.f;
    for(int i=0;i<CC;i++)
      for(int k=0;k<3;k++)
        a += Wr[i*3+k]*Xs[i*34+nbase+n+k];
    vals[r]=a;
  }
#endif
  #pragma unroll
  for(int r=0;r<8;r++){
    int ch=mt*16 + r + 8*half;          // D layout: VGPR r -> M=r (+8 for hi lanes)
    int pos=l0+nbase+n;
    size_t idx=base+(size_t)ch*CL+pos;
    float v=vals[r]+bias[ch];
    if(mode==MODE_GELU){
      out[idx]=gelu_f(v);
    } else if(mode==MODE_ODE2){
      float m2=v*g1beta[ch]+g1beta[CHID+ch];
      float yc=yin[idx];
      if(kbuf) yc += alpha*kbuf[idx];
      float kv=silu_f(m2)-0.1f*yc;
      kout[idx]=kv;
      yacc[idx]+=wfrac*dt*kv;
    } else { // MODE_GATE
      float g=sigmoid_f(v);
      float cur=Xs[ch*34 + (pos-l0) + 1];
      out[idx]=cur+g*detail[idx];
    }
  }
}

// --------------------------------------------------- K=1 attention (64<->16)
__global__ void attn_kernel(const float* __restrict__ cur, float* __restrict__ detail,
   const float* __restrict__ w1,const float* __restrict__ b1,
   const float* __restrict__ w2,const float* __restrict__ b2,int li){
  int gid=blockIdx.x*256+threadIdx.x;
  if(gid>=CB*CL) return;
  int b=gid>>11, p=gid&2047;
  float h[16];
  for(int o=0;o<16;o++) h[o]=b1[li*16+o];
  size_t base=(size_t)b*NPER+p;
  for(int c=0;c<CC;c++){
    float cv=cur[base+(size_t)c*CL];
    #pragma unroll
    for(int o=0;o<16;o++) h[o]+=w1[li*1024+o*CC+c]*cv;
  }
  for(int o=0;o<16;o++) h[o]=gelu_f(h[o]);
  for(int c=0;c<CC;c++){
    float a=b2[li*CC+c];
    #pragma unroll
    for(int o=0;o<16;o++) a+=w2[li*1024+c*16+o]*h[o];
    float s=sigmoid_f(a);
    size_t idx=base+(size_t)c*CL;
    detail[idx]=detail[idx]*(1.0f+s);     // d*a + d
  }
}

// --------------------------------------------------------- energy rescale
__global__ void energy_reduce_kernel(const float* __restrict__ c,
    const float* __restrict__ y,float* __restrict__ scale){
  int b=blockIdx.x;
  size_t base=(size_t)b*NPER;
  float s1=0.f,s2=0.f;
  for(int i=threadIdx.x;i<NPER;i+=256){
    float a=c[base+i], d=y[base+i];
    s1+=a*a; s2+=d*d;
  }
  __shared__ float r1[256],r2[256];
  r1[threadIdx.x]=s1; r2[threadIdx.x]=s2; __syncthreads();
  for(int off=128;off>0;off>>=1){
    if(threadIdx.x<off){ r1[threadIdx.x]+=r1[threadIdx.x+off]; r2[threadIdx.x]+=r2[threadIdx.x+off]; }
    __syncthreads();
  }
  if(threadIdx.x==0){
    float ein=r1[0]*(1.0f/NPER), eout=r2[0]*(1.0f/NPER);
    scale[b]=sqrtf(ein/(eout+1e-8f));
  }
}

__global__ void energy_apply_kernel(const float* __restrict__ y,
    const float* __restrict__ scale,float* __restrict__ dst){
  int gid=blockIdx.x*256+threadIdx.x;
  if(gid<NTOT) dst[gid]=y[gid]*scale[gid>>17];
}

__global__ void copy_scalar_kernel(const float* __restrict__ src,float* __restrict__ dst){
  if(blockIdx.x==0 && threadIdx.x==0) dst[0]=src[0];
}

// ---------------------------------------------------------------------------
extern "C" void kernel_launch(void* const* d_in, const int* in_sizes, int n_in,
                              void* d_out, int out_size, void* d_ws, size_t ws_size,
                              hipStream_t stream){
  const float* x       =(const float*)d_in[0];
  const float* dy_w1   =(const float*)d_in[1];
  const float* dy_b1   =(const float*)d_in[2];
  const float* dy_wg1  =(const float*)d_in[3];
  const float* dy_bg1  =(const float*)d_in[4];
  const float* dy_wg2  =(const float*)d_in[5];
  const float* dy_bg2  =(const float*)d_in[6];
  const float* ode_c1w =(const float*)d_in[7];
  const float* ode_c1b =(const float*)d_in[8];
  const float* ode_c2w =(const float*)d_in[9];
  const float* ode_c2b =(const float*)d_in[10];
  const float* ode_wt  =(const float*)d_in[11];
  const float* ode_bt  =(const float*)d_in[12];
  const float* ode_wm  =(const float*)d_in[13];
  const float* ode_bm  =(const float*)d_in[14];
  const float* ode_emb =(const float*)d_in[15];
  const float* gate_w  =(const float*)d_in[16];
  const float* gate_b  =(const float*)d_in[17];
  const float* attn1_w =(const float*)d_in[18];
  const float* attn1_b =(const float*)d_in[19];
  const float* attn2_w =(const float*)d_in[20];
  const float* attn2_b =(const float*)d_in[21];

  float* ws=(float*)d_ws;
  size_t Nf=(size_t)NTOT;
  float* bufA=ws;        float* bufB=ws+Nf;
  float* D1=ws+2*Nf;     float* D2=ws+3*Nf;    float* D3=ws+4*Nf;
  float* YACC=ws+5*Nf;   float* Hb=ws+6*Nf;    float* Kb=ws+7*Nf;   float* CUR=ws+8*Nf;
  float* sm=ws+9*Nf;
  float* stat=sm;        float* lo=sm+512;     float* hi=sm+576;    float* ortho=sm+640;
  float* tsb=sm+648;     float* rowmin=sm+704; float* rowmax=sm+1216;
  float* rowsum=sm+1728; float* gstat=sm+3776; float* g1beta=sm+3840; float* scale=sm+3968;

  size_t bytes=Nf*sizeof(float);
  hipMemcpyAsync(bufA,x,bytes,hipMemcpyDeviceToDevice,stream);

  // ---- wavelet decomposition (3 levels, dynamic filters) ----
  float* approx=bufA; float* tmp=bufB;
  float* dets[3]={D1,D2,D3};
  for(int lvl=0;lvl<NLEV;lvl++){
    row_mean_kernel<<<CB*CC,256,0,stream>>>(approx,stat);
    dywan_kernel<<<1,256,0,stream>>>(stat,dy_w1,dy_b1,dy_wg1,dy_bg1,dy_wg2,dy_bg2,lo,hi,ortho);
    filter_kernel<<<CB*CC,256,0,stream>>>(approx,lo,hi,tmp,dets[lvl]);
    float* t2=approx; approx=tmp; tmp=t2;
  }
  float* coeff0=approx; float* freeBuf=tmp;

  float* outp=(float*)d_out;
  float* cbufs[4]={coeff0,D1,D2,D3};
  float* dests[4]={CUR,outp+Nf,outp+2*Nf,outp+3*Nf};
  const float afr[4]={0.f,0.5f,0.5f,1.f};
  const float tfr[4]={0.f,0.5f,0.5f,1.f};
  const float wfr[4]={1.f/6.f,2.f/6.f,2.f/6.f,1.f/6.f};

  // ---- RK4 ODE evolution per level ----
  for(int lvl=0;lvl<NLEV+1;lvl++){
    float* cb=cbufs[lvl];
    tg_minmax_kernel<<<CB*CC,256,0,stream>>>(cb,rowmin,rowmax);
    tg_global_kernel<<<1,256,0,stream>>>(rowmin,rowmax,gstat);
    tg_sums_kernel<<<CB*CC,256,0,stream>>>(cb,gstat,rowsum);
    tg_final_kernel<<<1,256,0,stream>>>(rowsum,tsb);

    float* y=freeBuf; float* ya=YACC;
    hipMemcpyAsync(y,cb,bytes,hipMemcpyDeviceToDevice,stream);
    const float* c1w=ode_c1w+(size_t)lvl*CHID*CC*3;
    const float* c1b=ode_c1b+lvl*CHID;
    const float* c2w=ode_c2w+(size_t)lvl*CHID*CHID*3;
    const float* c2b=ode_c2b+lvl*CHID;
    for(int j=0;j<NSTEP;j++){
      hipMemcpyAsync(ya,y,bytes,hipMemcpyDeviceToDevice,stream);
      for(int sub=0;sub<4;sub++){
        const float* kprev=(sub==0)?nullptr:Kb;
        modvec_kernel<<<1,128,0,stream>>>(tsb,j,tfr[sub],ode_wt,ode_bt,ode_wm,ode_bm,ode_emb,lvl,g1beta);
        conv3_kernel<<<CB*64,256,0,stream>>>(y,kprev,tsb,j,afr[sub],c1w,c1b,
            MODE_GELU,nullptr,nullptr,nullptr,0.f,nullptr,nullptr,Hb);
        conv3_kernel<<<CB*64,256,0,stream>>>(Hb,kprev,tsb,j,afr[sub],c2w,c2b,
            MODE_ODE2,g1beta,y,ya,wfr[sub],Kb,nullptr,nullptr);
      }
      float* t2=y; y=ya; ya=t2;
    }
    energy_reduce_kernel<<<CB,256,0,stream>>>(cb,y,scale);
    energy_apply_kernel<<<(NTOT+255)/256,256,0,stream>>>(y,scale,dests[lvl]);
  }

  // ---- reconstruction: i = 2, 1, 0 ----
  // i=2: no attention
  conv3_kernel<<<CB*64,256,0,stream>>>(CUR,nullptr,nullptr,0,0.f,
      gate_w+2*(size_t)CC*CC*3,gate_b+2*CC,
      MODE_GATE,nullptr,nullptr,nullptr,0.f,nullptr,outp+3*Nf,Hb);
  // i=1
  attn_kernel<<<(CB*CL+255)/256,256,0,stream>>>(Hb,outp+2*Nf,attn1_w,attn1_b,attn2_w,attn2_b,1);
  conv3_kernel<<<CB*64,256,0,stream>>>(Hb,nullptr,nullptr,0,0.f,
      gate_w+1*(size_t)CC*CC*3,gate_b+1*CC,
      MODE_GATE,nullptr,nullptr,nullptr,0.f,nullptr,outp+2*Nf,Kb);
  // i=0
  attn_kernel<<<(CB*CL+255)/256,256,0,stream>>>(Kb,outp+1*Nf,attn1_w,attn1_b,attn2_w,attn2_b,0);
  conv3_kernel<<<CB*64,256,0,stream>>>(Kb,nullptr,nullptr,0,0.f,
      gate_w,gate_b,
      MODE_GATE,nullptr,nullptr,nullptr,0.f,nullptr,outp+1*Nf,outp);

  copy_scalar_kernel<<<1,32,0,stream>>>(ortho,outp+4*Nf);
}